// adaptiveFitNet_75849122448021
// MI455X (gfx1250) — compile-verified
//
#include <hip/hip_runtime.h>

typedef __attribute__((ext_vector_type(2))) float v2f;
typedef __attribute__((ext_vector_type(4))) float v4f;
typedef __attribute__((ext_vector_type(8))) float v8f;

#define THRC 0.15f

__device__ __forceinline__ float waveSum(float v) {
#pragma unroll
  for (int o = 16; o > 0; o >>= 1) v += __shfl_xor(v, o, 32);
  return v;
}
__device__ __forceinline__ float waveMax(float v) {
#pragma unroll
  for (int o = 16; o > 0; o >>= 1) v = fmaxf(v, __shfl_xor(v, o, 32));
  return v;
}

// ---------------------------------------------------------------------------
// K1: w[b,c] = mean over HW of g[b,c,:,:]   (one wave32 per channel)
// HW % 4 == 0 implies every channel base is 16B-aligned -> b128 path.
// ---------------------------------------------------------------------------
template <int HW>
__global__ void __launch_bounds__(256)
grad_weights_kernel(const float* __restrict__ g, float* __restrict__ w, int nCh) {
  const int waveId = (blockIdx.x * blockDim.x + threadIdx.x) >> 5;
  const int lane   = threadIdx.x & 31;
  if (waveId >= nCh) return;
  const float* src = g + (size_t)waveId * HW;
  float s = 0.f;
  if constexpr ((HW & 3) == 0) {
    const v4f* src4 = (const v4f*)src;   // 16B-aligned: waveId*HW*4 % 16 == 0
    for (int i = lane; i < (HW >> 2); i += 32) {
      const v4f v = __builtin_nontemporal_load(src4 + i);
      s += (v.x + v.y) + (v.z + v.w);
    }
  } else {
    for (int i = lane; i < HW; i += 32) s += __builtin_nontemporal_load(src + i);
  }
  s = waveSum(s);
  if (lane == 0) w[waveId] = s / (float)HW;
}

// ---------------------------------------------------------------------------
// K2: fused channel reductions via V_WMMA_F32_16X16X4_F32.
//   D row0 = cam   = sum_c w[c]*Tf[c,n]
//   D row1 = meanT = (1/C) * sum_c Tf[c,n]
//   D row2 = meanS = (1/C) * sum_c Sf[c,n]
// One wave per (batch, 16-wide spatial tile); EXEC all-ones by construction.
// B per iter (K step = 2 channels):
//   VGPR0: lanes 0-15 = Tf[c]   (K=0), lanes 16-31 = Sf[c]   (K=2)
//   VGPR1: lanes 0-15 = Tf[c+1] (K=1), lanes 16-31 = Sf[c+1] (K=3)
// A (16x4 f32): lane0 = {w[c],w[c+1]}, lane1 = {1/C,1/C},
//   lane18 (M=2,K=2..3) = {1/C,1/C}, all other lanes zero.
// ---------------------------------------------------------------------------
template <int C, int HW>
__global__ void __launch_bounds__(32)
cam_wmma_kernel(const float* __restrict__ Tf, const float* __restrict__ Sf,
                const float* __restrict__ w,
                float* __restrict__ cam, float* __restrict__ meanT,
                float* __restrict__ meanS) {
  constexpr int TILES = (HW + 15) / 16;
  constexpr float INVC = 1.0f / (float)C;

  const int wave = blockIdx.x;         // exactly B*TILES single-wave blocks
  const int lane = threadIdx.x;        // 0..31
  const int b    = wave / TILES;
  const int n0   = (wave % TILES) * 16;
  const int nl   = lane & 15;
  const int half = lane >> 4;          // 0 -> Tf rows (K=0,1), 1 -> Sf rows (K=2,3)

  int n = n0 + nl;
  if (n >= HW) n = HW - 1;             // clamp: dup loads, results discarded
  const float* src  = (half ? Sf : Tf) + (size_t)b * C * HW + n;
  const float* wrow = w + (size_t)b * C;

  const bool  isW    = (lane == 0);
  const float aconst = (lane == 1 || lane == 18) ? INVC : 0.0f;

  v8f acc = {};
#pragma unroll 8
  for (int c = 0; c < C; c += 2) {
    const float w0 = wrow[c];
    const float w1 = wrow[c + 1];
    v2f a, bm;
    a.x  = isW ? w0 : aconst;
    a.y  = isW ? w1 : aconst;
    bm.x = __builtin_nontemporal_load(src + (size_t)c * HW);
    bm.y = __builtin_nontemporal_load(src + (size_t)(c + 1) * HW);
    acc = __builtin_amdgcn_wmma_f32_16x16x4_f32(false, a, false, bm,
                                                (short)0, acc, false, false);
  }

  if (lane < 16 && (n0 + nl) < HW) {   // D: VGPR r = row M=r for lanes 0-15
    const size_t o = (size_t)b * HW + n0 + nl;
    cam[o]   = acc[0];
    meanT[o] = acc[1];
    meanS[o] = acc[2];
  }
}

// ---------------------------------------------------------------------------
// K3: per-batch stats on raw cam: stats[2b] = max(cam), stats[2b+1] = mean(relu(cam))
// ---------------------------------------------------------------------------
__global__ void __launch_bounds__(256)
cam_stats_kernel(const float* __restrict__ cam, float* __restrict__ stats, int HW) {
  __shared__ float smax[8], ssum[8];
  const int b = blockIdx.x;
  const float* src = cam + (size_t)b * HW;
  float mx = -3.4e38f, sm = 0.f;
  for (int i = threadIdx.x; i < HW; i += blockDim.x) {
    const float v = src[i];
    mx = fmaxf(mx, v);
    sm += fmaxf(v, 0.f);
  }
  mx = waveMax(mx);
  sm = waveSum(sm);
  const int wid = threadIdx.x >> 5, lane = threadIdx.x & 31;
  if (lane == 0) { smax[wid] = mx; ssum[wid] = sm; }
  __syncthreads();
  if (threadIdx.x == 0) {
    float M = smax[0], S = ssum[0];
    for (int i = 1; i < (int)(blockDim.x >> 5); ++i) {
      M = fmaxf(M, smax[i]);
      S += ssum[i];
    }
    stats[2 * b]     = M;
    stats[2 * b + 1] = S / (float)HW;
  }
}

// ---------------------------------------------------------------------------
// K4: masked embeddings. mask = cam > |max*0.15| (equiv. to clip-then-compare)
// ---------------------------------------------------------------------------
__global__ void __launch_bounds__(256)
emb_kernel(const float* __restrict__ cam, const float* __restrict__ meanT,
           const float* __restrict__ meanS, const float* __restrict__ stats,
           float* __restrict__ embT, float* __restrict__ embS,
           int HW, int total) {
  const int i = blockIdx.x * blockDim.x + threadIdx.x;
  if (i >= total) return;
  const int b = i / HW;
  const float t = fabsf(stats[2 * b] * THRC);
  const bool  m = cam[i] > t;
  embT[i] = m ? meanT[i] : 0.0f;
  embS[i] = m ? meanS[i] : 0.0f;
}

// ---------------------------------------------------------------------------
// K5: fw = normalize([sum_b s0, sum_b s1])
// ---------------------------------------------------------------------------
__global__ void __launch_bounds__(32)
fw_kernel(const float* __restrict__ st0, const float* __restrict__ st1,
          float* __restrict__ out, int B) {
  const int lane = threadIdx.x;
  float s0 = (lane < B) ? st0[2 * lane + 1] : 0.f;
  float s1 = (lane < B) ? st1[2 * lane + 1] : 0.f;
  s0 = waveSum(s0);
  s1 = waveSum(s1);
  if (lane == 0) {
    const float inv = 1.0f / (s0 + s1);
    out[0] = s0 * inv;
    out[1] = s1 * inv;
  }
}

extern "C" void kernel_launch(void* const* d_in, const int* in_sizes, int n_in,
                              void* d_out, int out_size, void* d_ws, size_t ws_size,
                              hipStream_t stream) {
  (void)in_sizes; (void)n_in; (void)out_size; (void)ws_size;
  constexpr int B = 32, C0 = 512, C1 = 1024, HW0 = 38 * 38, HW1 = 19 * 19;

  const float* Tf0 = (const float*)d_in[0];
  const float* Tf1 = (const float*)d_in[1];
  const float* Sf0 = (const float*)d_in[2];
  const float* Sf1 = (const float*)d_in[3];
  const float* g0  = (const float*)d_in[4];
  const float* g1  = (const float*)d_in[5];

  // workspace carve-up (floats)
  float* f    = (float*)d_ws;
  float* w0   = f;                 // B*C0
  float* w1   = w0 + B * C0;       // B*C1
  float* cam0 = w1 + B * C1;       // B*HW0
  float* mT0  = cam0 + B * HW0;    // B*HW0
  float* mS0  = mT0 + B * HW0;     // B*HW0
  float* cam1 = mS0 + B * HW0;     // B*HW1
  float* mT1  = cam1 + B * HW1;    // B*HW1
  float* mS1  = mT1 + B * HW1;     // B*HW1
  float* st0  = mS1 + B * HW1;     // 2*B
  float* st1  = st0 + 2 * B;       // 2*B

  // output layout: fw(2) | embT0 | embT1 | embS0 | embS1
  float* out   = (float*)d_out;
  float* embT0 = out + 2;
  float* embT1 = embT0 + B * HW0;
  float* embS0 = embT1 + B * HW1;
  float* embS1 = embS0 + B * HW0;

  // K1: gradient channel means (one wave per channel, 8 waves/block)
  const int nCh0 = B * C0, nCh1 = B * C1;
  grad_weights_kernel<HW0><<<(nCh0 + 7) / 8, 256, 0, stream>>>(g0, w0, nCh0);
  grad_weights_kernel<HW1><<<(nCh1 + 7) / 8, 256, 0, stream>>>(g1, w1, nCh1);

  // K2: fused cam + channel means via f32 WMMA (one wave per block -> EXEC full)
  constexpr int tiles0 = (HW0 + 15) / 16, tiles1 = (HW1 + 15) / 16;
  cam_wmma_kernel<C0, HW0><<<B * tiles0, 32, 0, stream>>>(Tf0, Sf0, w0,
                                                          cam0, mT0, mS0);
  cam_wmma_kernel<C1, HW1><<<B * tiles1, 32, 0, stream>>>(Tf1, Sf1, w1,
                                                          cam1, mT1, mS1);

  // K3: per-batch max / score
  cam_stats_kernel<<<B, 256, 0, stream>>>(cam0, st0, HW0);
  cam_stats_kernel<<<B, 256, 0, stream>>>(cam1, st1, HW1);

  // K4: masked embeddings
  emb_kernel<<<(B * HW0 + 255) / 256, 256, 0, stream>>>(cam0, mT0, mS0, st0,
                                                        embT0, embS0, HW0, B * HW0);
  emb_kernel<<<(B * HW1 + 255) / 256, 256, 0, stream>>>(cam1, mT1, mS1, st1,
                                                        embT1, embS1, HW1, B * HW1);

  // K5: level weights
  fw_kernel<<<1, 32, 0, stream>>>(st0, st1, out, B);
}